// CapsuleRouting_26534307955118
// MI455X (gfx1250) — compile-verified
//
#include <hip/hip_runtime.h>

// Capsule routing, fully fused, MI455X (gfx1250, wave32).
// One workgroup (256 thr = 8 waves) per 4 consecutive spatial sites.
// u is recomputed via V_WMMA_F32_16X16X4_F32 each pass; never hits HBM.
//
// Shapes: p (4,144,16,14,14) f32; a (4,144,14,14) f32 (UNUSED by reference);
//         W (144,4,4,32) f32. Out: v (4,32,16,14,14) ++ a_out (4,32,14,14).

typedef float v2f __attribute__((ext_vector_type(2)));
typedef float v4f __attribute__((ext_vector_type(4)));
typedef float v8f __attribute__((ext_vector_type(8)));

#define NB   144   // B_ capsules in
#define NC   32    // C_ capsules out
#define HW   196   // 14*14
#define NS   4     // sites per workgroup
#define EPSF 1e-5f

// LDS partition (floats):
//  pL [4][144][16]  = 9216   staged poses for 4 sites
//  rS [4][144][32]  = 18432  routing logits
//  cS [144][32][4]  = 18432  softmax coeffs, site-fastest for ds_load_b128
//  sS [4][32][16]   = 2048   weighted sums
//  vT [32][16][4]   = 2048   v, site-fastest for ds_load_b64 in pass B
// total 50176 floats = 200704 bytes (<= 320KB WGP LDS)
#define OFF_R  9216
#define OFF_C  (OFF_R + 4*NB*NC)
#define OFF_S  (OFF_C + NB*NC*4)
#define OFF_V  (OFF_S + NS*NC*16)
#define LDS_FLOATS (OFF_V + NC*16*NS)

__launch_bounds__(256)
__global__ void caps_route_kernel(const float* __restrict__ p,
                                  const float* __restrict__ W,
                                  float* __restrict__ out) {
  extern __shared__ float smem[];
  float* pL = smem;
  float* rS = smem + OFF_R;
  float* cS = smem + OFF_C;
  float* sS = smem + OFF_S;
  float* vT = smem + OFF_V;

  const int tid = threadIdx.x;
  const int wg  = blockIdx.x;       // 0..195
  const int b   = wg / 49;          // batch
  const int hw0 = (wg % 49) * 4;    // first of 4 consecutive hw sites

  // ---- stage p: coalesced float4 over 4 sites (hw0..hw0+3, 16B aligned) ----
  for (int pair = tid; pair < NB * 16; pair += 256) {
    const float4 val =
        *(const float4*)(p + ((size_t)b * NB * 16 + pair) * HW + hw0);
    const int Bc = pair >> 4, pp = pair & 15;
    pL[(0 * NB + Bc) * 16 + pp] = val.x;
    pL[(1 * NB + Bc) * 16 + pp] = val.y;
    pL[(2 * NB + Bc) * 16 + pp] = val.z;
    pL[(3 * NB + Bc) * 16 + pp] = val.w;
  }
  // r = 0; c = softmax(0) = 1/32 (lets iteration 0 reuse the generic pass)
  for (int i = tid; i < 4 * NB * NC; i += 256) {
    rS[i] = 0.0f;
    cS[i] = 0.03125f;
  }
  __syncthreads();

  // ---- wave -> N-tile assignment ----
  const int  lane  = tid & 31;
  const int  wv    = tid >> 5;                 // tile n = wave id, 0..7
  const int  k_n   = wv >> 1;                  // k index of this tile
  const int  Ccol  = (wv & 1) * 16 + (lane & 15);  // C column owned by lane
  const int  m15   = lane & 15;
  const int  siteA = m15 >> 2;                 // A-operand row -> (site,i)
  const int  iA    = m15 & 3;
  const bool lo    = lane < 16;
  // A 16x4 f32 layout: v0 = K{0|2}, v1 = K{1|3} split on lane halves
  const int  jA    = lo ? 0 : 2;
  // B 4x16 f32 layout: v0 = K{0|1}, v1 = K{2|3} split on lane halves
  const int  jB    = lo ? 0 : 1;

  const int aBase = (siteA * NB) * 16 + iA * 4 + jA;  // + Bc*16 per capsule
  const size_t wStride = (size_t)4 * 4 * NC;          // W per-B stride

  for (int it = 0; it < 3; ++it) {
    // ======== Pass A: s[site,C,PP] = sum_B c[site,B,C] * u ========
    float acc[8];
#pragma unroll
    for (int r = 0; r < 8; ++r) acc[r] = 0.0f;

    for (int Bc = 0; Bc < NB; ++Bc) {
      const v2f A = *(const v2f*)&pL[aBase + Bc * 16];
      v2f Bm;
      Bm.x = W[(size_t)Bc * wStride + ((jB    ) * 4 + k_n) * NC + Ccol];
      Bm.y = W[(size_t)Bc * wStride + ((jB + 2) * 4 + k_n) * NC + Ccol];
      v8f c0 = {};
      const v8f d = __builtin_amdgcn_wmma_f32_16x16x4_f32(
          false, A, false, Bm, (short)0, c0, false, false);
      const v4f cw = *(const v4f*)&cS[((size_t)Bc * NC + Ccol) * 4];
      const float cA = lo ? cw.x : cw.z;   // sites 0/2
      const float cB = lo ? cw.y : cw.w;   // sites 1/3
#pragma unroll
      for (int r = 0; r < 4; ++r) acc[r]     += d[r]     * cA;
#pragma unroll
      for (int r = 0; r < 4; ++r) acc[4 + r] += d[4 + r] * cB;
    }
    // each wave exclusively owns (C-half, k) columns -> plain stores
#pragma unroll
    for (int r = 0; r < 8; ++r) {
      const int m = r + (lo ? 0 : 8);
      const int site = m >> 2, ii = m & 3;
      sS[(site * NC + Ccol) * 16 + ii * 4 + k_n] = acc[r];
    }
    __syncthreads();

    // ======== squash: v = sqrt(n2)/(1+n2) * s, n2 = sum s^2 + EPS ========
    if (tid < 128) {
      const int C = tid & 31, site = tid >> 5;
      float sv[16];
      float n2 = EPSF;
#pragma unroll
      for (int pp = 0; pp < 16; ++pp) {
        sv[pp] = sS[(site * NC + C) * 16 + pp];
        n2 += sv[pp] * sv[pp];
      }
      const float f = sqrtf(n2) / (1.0f + n2);
#pragma unroll
      for (int pp = 0; pp < 16; ++pp) vT[(C * 16 + pp) * 4 + site] = f * sv[pp];
    }
    __syncthreads();

    if (it < 2) {
      // ======== Pass B: r[site,B,C] += sum_PP u * v ========
      for (int Bc = 0; Bc < NB; ++Bc) {
        const v2f A = *(const v2f*)&pL[aBase + Bc * 16];
        v2f Bm;
        Bm.x = W[(size_t)Bc * wStride + ((jB    ) * 4 + k_n) * NC + Ccol];
        Bm.y = W[(size_t)Bc * wStride + ((jB + 2) * 4 + k_n) * NC + Ccol];
        v8f c0 = {};
        const v8f d = __builtin_amdgcn_wmma_f32_16x16x4_f32(
            false, A, false, Bm, (short)0, c0, false, false);
        float tA = 0.0f, tB = 0.0f;
#pragma unroll
        for (int ii = 0; ii < 4; ++ii) {
          // v pairs for (site, site+1) at this lane's (C, PP=ii*4+k_n)
          const v2f vv = *(const v2f*)&vT[(Ccol * 16 + ii * 4 + k_n) * 4 +
                                          (lo ? 0 : 2)];
          tA += d[ii]     * vv.x;
          tB += d[4 + ii] * vv.y;
        }
        const int sA = lo ? 0 : 2;
        atomicAdd(&rS[((sA    ) * NB + Bc) * NC + Ccol], tA);  // ds_add_f32
        atomicAdd(&rS[((sA + 1) * NB + Bc) * NC + Ccol], tB);
      }
      __syncthreads();

      // ======== softmax over C per (site,B) row ========
      for (int row = tid; row < 4 * NB; row += 256) {
        const float* rr = &rS[row * NC];
        const int site = row / NB, Bc = row % NB;
        float mx = -1e30f;
        for (int C = 0; C < NC; ++C) mx = fmaxf(mx, rr[C]);
        float sum = 0.0f;
        for (int C = 0; C < NC; ++C) sum += __expf(rr[C] - mx);
        const float inv = 1.0f / sum;
        for (int C = 0; C < NC; ++C)
          cS[((size_t)Bc * NC + C) * 4 + site] = __expf(rr[C] - mx) * inv;
      }
      __syncthreads();
    }
  }

  // ======== outputs: v (4,32,16,14,14) then a_out (4,32,14,14) ========
  for (int idx = tid; idx < NS * NC * 16; idx += 256) {
    const int site = idx & 3, rem = idx >> 2;
    const int PP = rem & 15, C = rem >> 4;
    out[(((size_t)b * NC + C) * 16 + PP) * HW + hw0 + site] =
        vT[(C * 16 + PP) * 4 + site];
  }
  if (tid < 128) {
    const int site = tid & 3, C = tid >> 2;
    float n2 = EPSF;
#pragma unroll
    for (int PP = 0; PP < 16; ++PP) {
      const float vv = vT[(C * 16 + PP) * 4 + site];
      n2 += vv * vv;
    }
    out[(size_t)4 * NC * 16 * HW + ((size_t)b * NC + C) * HW + hw0 + site] =
        sqrtf(n2);
  }
}

extern "C" void kernel_launch(void* const* d_in, const int* in_sizes, int n_in,
                              void* d_out, int out_size, void* d_ws,
                              size_t ws_size, hipStream_t stream) {
  (void)in_sizes; (void)n_in; (void)d_ws; (void)ws_size; (void)out_size;
  const float* p = (const float*)d_in[0];
  // d_in[1] = a : unused by the reference computation
  const float* W = (const float*)d_in[2];
  float* out = (float*)d_out;

  const size_t ldsBytes = (size_t)LDS_FLOATS * sizeof(float);  // 200704 B
  hipLaunchKernelGGL(caps_route_kernel, dim3(196), dim3(256), ldsBytes, stream,
                     p, W, out);
}